// Net_21852793602541
// MI455X (gfx1250) — compile-verified
//
#include <hip/hip_runtime.h>
#include <hip/hip_bf16.h>
#include <stdint.h>

// ---------------------------------------------------------------------------
// Graph-circuit net for MI455X (gfx1250, wave32).
//  - graph_prep: device-side BFS depth + per-neuron ordered in-lists.
//  - conv_wave (x48): depth-gated 5x5 sparse-channel convs, LDS-tiled.
//  - gather:   depth-masked channel-major gather for neuron 255's FC input.
//  - fc1:      f32 GEMM on matrix cores (V_WMMA_F32_16X16X4_F32) + bias+ReLU.
//  - fc2:      small FC + log-softmax.
// ---------------------------------------------------------------------------

#define N_NEURONS 256
#define N_EDGES   1024
#define BATCH     128
#define HWD       28
#define PIX       (HWD * HWD)      // 784
#define FC_HID    200
#define N_CLS     10
#define INF_D     (1 << 20)
#define WAVE_MAX  48               // upper bound on BFS depth (typ. ~6)

typedef float v2f __attribute__((ext_vector_type(2)));
typedef float v8f __attribute__((ext_vector_type(8)));

// ---------------------------------------------------------------------------
// 1) Build per-neuron in-lists (edge order) and BFS depths (Bellman-Ford in
//    LDS; 256 threads, 256 relaxation rounds => exact BFS distance).
// ---------------------------------------------------------------------------
__global__ void graph_prep_kernel(const int* __restrict__ src,
                                  const int* __restrict__ tgt,
                                  int* __restrict__ idx,
                                  int* __restrict__ cnt,
                                  int* __restrict__ depth,
                                  int cmax) {
  const int v = threadIdx.x;   // 0..255
  int c = 0;
  for (int e = 0; e < N_EDGES; ++e) {
    if (tgt[e] == v && c < cmax) { idx[v * cmax + c] = src[e]; ++c; }
  }
  cnt[v] = c;

  __shared__ int sd[N_NEURONS];
  sd[v] = (v == 0) ? 0 : INF_D;
  __syncthreads();
  for (int it = 0; it < N_NEURONS; ++it) {
    int best = INF_D;
    if (v != 0) {
      for (int j = 0; j < c; ++j) {
        int dv = sd[idx[v * cmax + j]] + 1;
        best = dv < best ? dv : best;
      }
    }
    __syncthreads();
    if (best < sd[v]) sd[v] = best;
    __syncthreads();
  }
  depth[v] = sd[v];
}

// ---------------------------------------------------------------------------
// 2) One BFS wave of convs. Block = (neuron v, batch-chunk of 4 images).
//    Active only when depth[v]==d. Input channel u contributes iff
//    depth[u] < d (snapshot semantics). d==0: neuron 0 reads image x with
//    weight channel 0 only. Padded 32x32 input tiles staged in LDS (16 KB).
// ---------------------------------------------------------------------------
__global__ void conv_wave_kernel(int d,
                                 const float* __restrict__ x,
                                 const float* __restrict__ cw,   // [256][cmax][25]
                                 const float* __restrict__ cb,   // [256]
                                 const int* __restrict__ idx,
                                 const int* __restrict__ cnt,
                                 const int* __restrict__ depth,
                                 float* __restrict__ outbuf,     // [256][128][784]
                                 int cmax) {
  const int v = blockIdx.x;
  if (v == N_NEURONS - 1) return;        // neuron 255 never runs a conv
  if (depth[v] != d) return;
  const int b0  = blockIdx.y * 4;
  const int tid = threadIdx.x;           // 256 threads

  __shared__ float s_in[4][32][32];
  __shared__ float s_w[32];

  float acc[13];
  #pragma unroll
  for (int k = 0; k < 13; ++k) acc[k] = 0.f;

  const int nc = (d == 0) ? 1 : cnt[v];
  for (int c = 0; c < nc; ++c) {
    const int  u    = (d == 0) ? 0 : idx[v * cmax + c];
    const bool live = (d == 0) || (depth[u] < d);
    if (!live) continue;                 // block-uniform: safe around barriers
    __syncthreads();                     // WAR guard vs previous channel
    if (tid < 25) s_w[tid] = cw[((size_t)v * cmax + c) * 25 + tid];
    const float* base = (d == 0) ? x : (outbuf + (size_t)u * BATCH * PIX);
    for (int t = tid; t < 4 * 32 * 32; t += 256) {
      const int img = t >> 10, rem = t & 1023, yy = rem >> 5, xx = rem & 31;
      const int iy = yy - 2, ix = xx - 2;
      float val = 0.f;
      if (iy >= 0 && iy < HWD && ix >= 0 && ix < HWD)
        val = base[(size_t)(b0 + img) * PIX + iy * HWD + ix];
      s_in[img][yy][xx] = val;
    }
    __syncthreads();
    #pragma unroll
    for (int k = 0; k < 13; ++k) {
      const int o = tid + (k << 8);
      if (o < 4 * PIX) {
        const int img = o / PIX, p = o % PIX, y = p / HWD, xc = p % HWD;
        float s = 0.f;
        #pragma unroll
        for (int dy = 0; dy < 5; ++dy)
          #pragma unroll
          for (int dx = 0; dx < 5; ++dx)
            s += s_in[img][y + dy][xc + dx] * s_w[dy * 5 + dx];
        acc[k] += s;
      }
    }
  }
  const float bb = cb[v];
  #pragma unroll
  for (int k = 0; k < 13; ++k) {
    const int o = tid + (k << 8);
    if (o < 4 * PIX) {
      const int img = o / PIX, p = o % PIX;
      const float r = acc[k] + bb;
      outbuf[((size_t)v * BATCH + (b0 + img)) * PIX + p] = r > 0.f ? r : 0.f;
    }
  }
}

// ---------------------------------------------------------------------------
// 3) Gather neuron 255's FC input, channel-major: comp[b][j*784+p], masked by
//    depth[u] < depth[255] (the snapshot at the final wave).
// ---------------------------------------------------------------------------
__global__ void gather_kernel(const float* __restrict__ outbuf,
                              const int* __restrict__ idx,
                              const int* __restrict__ depth,
                              float* __restrict__ comp,
                              int cmax, int deg_last) {
  const int total = BATCH * deg_last * PIX;
  const int t = blockIdx.x * blockDim.x + threadIdx.x;
  if (t >= total) return;
  const int p = t % PIX;
  const int j = (t / PIX) % deg_last;
  const int b = t / (PIX * deg_last);
  const int u = idx[(N_NEURONS - 1) * cmax + j];
  const int D = depth[N_NEURONS - 1];
  float val = 0.f;
  if (depth[u] < D) val = outbuf[((size_t)u * BATCH + b) * PIX + p];
  comp[t] = val;
}

// ---------------------------------------------------------------------------
// 4) FC1 on matrix cores: hidden = relu(comp[128xK] @ fc1_w^T[Kx200] + b).
//    One wave per 16x16 tile; V_WMMA_F32_16X16X4_F32, K-step 4.
//    A lane L: M=L%16, holds K = 2*(L/16)+{0,1}. Same split for B (KxN).
//    D lane L: N=L%16, acc[r] -> M = r + 8*(L/16).
// ---------------------------------------------------------------------------
__global__ void fc1_wmma_kernel(const float* __restrict__ comp,
                                const float* __restrict__ w,     // [200][K]
                                const float* __restrict__ bias,  // [200]
                                float* __restrict__ hidden,      // [128][200]
                                int K) {
  const int lane = threadIdx.x;          // 0..31, one wave per block
  const int half = lane >> 4;
  const int lr   = lane & 15;
  const int m = blockIdx.x * 16 + lr;            // < 128 always
  const int n = blockIdx.y * 16 + lr;
  const int nsafe = (n < FC_HID) ? n : 0;        // clamp: no EXEC divergence
  const float* arow = comp + (size_t)m * K;
  const float* brow = w + (size_t)nsafe * K;

  v8f acc = {};
  for (int k = 0; k < K; k += 4) {
    const int kk = k + half * 2;
    v2f a; a.x = arow[kk]; a.y = arow[kk + 1];
    v2f b; b.x = brow[kk]; b.y = brow[kk + 1];
    acc = __builtin_amdgcn_wmma_f32_16x16x4_f32(
        /*neg_a=*/false, a, /*neg_b=*/false, b,
        /*c_mod=*/(short)0, acc, /*reuse_a=*/false, /*reuse_b=*/false);
  }

  if (n < FC_HID) {
    const float bn = bias[n];
    #pragma unroll
    for (int r = 0; r < 8; ++r) {
      const int mrow = blockIdx.x * 16 + half * 8 + r;
      const float vv = acc[r] + bn;
      hidden[mrow * FC_HID + n] = vv > 0.f ? vv : 0.f;
    }
  }
}

// ---------------------------------------------------------------------------
// 5) FC2 + log-softmax. One block per batch row.
// ---------------------------------------------------------------------------
__global__ void fc2_lsm_kernel(const float* __restrict__ hidden,
                               const float* __restrict__ w2,  // [10][200]
                               const float* __restrict__ b2,
                               float* __restrict__ out) {
  const int b = blockIdx.x, tid = threadIdx.x;
  __shared__ float lg[N_CLS];
  if (tid < N_CLS) {
    float s = b2[tid];
    const float* h  = hidden + b * FC_HID;
    const float* wr = w2 + tid * FC_HID;
    for (int j = 0; j < FC_HID; ++j) s += h[j] * wr[j];
    lg[tid] = s;
  }
  __syncthreads();
  if (tid < N_CLS) {
    float m = lg[0];
    for (int c = 1; c < N_CLS; ++c) m = fmaxf(m, lg[c]);
    float sum = 0.f;
    for (int c = 0; c < N_CLS; ++c) sum += expf(lg[c] - m);
    out[b * N_CLS + tid] = lg[tid] - m - logf(sum);
  }
}

// ---------------------------------------------------------------------------
extern "C" void kernel_launch(void* const* d_in, const int* in_sizes, int n_in,
                              void* d_out, int out_size, void* d_ws, size_t ws_size,
                              hipStream_t stream) {
  (void)n_in; (void)out_size; (void)ws_size;
  const float* x      = (const float*)d_in[0];
  const int*   src    = (const int*)d_in[1];
  const int*   tgt    = (const int*)d_in[2];
  const float* conv_w = (const float*)d_in[3];
  const float* conv_b = (const float*)d_in[4];
  const float* fc1_w  = (const float*)d_in[5];
  const float* fc1_b  = (const float*)d_in[6];
  const float* fc2_w  = (const float*)d_in[7];
  const float* fc2_b  = (const float*)d_in[8];

  const int cmax     = in_sizes[3] / (N_NEURONS * 25);
  const int Kfc      = in_sizes[5] / FC_HID;   // deg_last * 784
  const int deg_last = Kfc / PIX;

  char* p = (char*)d_ws;
  int* idx   = (int*)p; p += (size_t)N_NEURONS * cmax * sizeof(int);
  int* cnt   = (int*)p; p += (size_t)N_NEURONS * sizeof(int);
  int* depth = (int*)p; p += (size_t)N_NEURONS * sizeof(int);
  p = (char*)(((uintptr_t)p + 255) & ~(uintptr_t)255);
  float* outbuf = (float*)p; p += (size_t)N_NEURONS * BATCH * PIX * sizeof(float);
  float* comp   = (float*)p; p += (size_t)BATCH * Kfc * sizeof(float);
  float* hidden = (float*)p; p += (size_t)BATCH * FC_HID * sizeof(float);

  graph_prep_kernel<<<1, N_NEURONS, 0, stream>>>(src, tgt, idx, cnt, depth, cmax);

  for (int d = 0; d < WAVE_MAX; ++d) {
    conv_wave_kernel<<<dim3(N_NEURONS, BATCH / 4), 256, 0, stream>>>(
        d, x, conv_w, conv_b, idx, cnt, depth, outbuf, cmax);
  }

  const int total = BATCH * Kfc;
  gather_kernel<<<(total + 255) / 256, 256, 0, stream>>>(
      outbuf, idx, depth, comp, cmax, deg_last);

  fc1_wmma_kernel<<<dim3(BATCH / 16, (FC_HID + 15) / 16), 32, 0, stream>>>(
      comp, fc1_w, fc1_b, hidden, Kfc);

  fc2_lsm_kernel<<<BATCH, 32, 0, stream>>>(hidden, fc2_w, fc2_b, (float*)d_out);
}